// DirectVolumeStratifiedFrontToBackRenderer_59992103190955
// MI455X (gfx1250) — compile-verified
//
#include <hip/hip_runtime.h>
#include <hip/hip_bf16.h>

typedef __attribute__((ext_vector_type(2))) float v2f;
typedef __attribute__((ext_vector_type(8))) float v8f;

#define IMG 256
#define NPTS 320
#define FOV_TAN 0.57735026918962576451f     /* tan(30 deg) */
#define GSTEP (2.0f / 255.0f)               /* linspace(-1,1,256) step */
#define DSTEP (4.0f / 319.0f)               /* linspace(2,6,320) step  */
#define HALFC 1.494140625f                  /* 255 * (3/256) * 0.5 = 765/512, exact */

// One wave = 16 pixels (one image row segment). Lane m (m=lane&15) owns pixel m.
// Lanes 0-15 handle samples 4t+{0,1}, lanes 16-31 handle samples 4t+{2,3} of their
// pixel -- exactly the VGPR ownership of both the A(16x4 f32) and B(4x16 f32)
// operands of V_WMMA_F32_16X16X4_F32 (A: v0=K0|K2, v1=K1|K3, M=lane%16; B: v0=row0|row2,
// v1=row1|row3, N=lane%16). D[m][m] then accumulates Sum_k w[m,k]*f[m,k] in full f32.
__global__ __launch_bounds__(256) void render_kernel(
    const float* __restrict__ vol, const float* __restrict__ Rm,
    const float* __restrict__ Tm, float* __restrict__ grayOut)
{
    __shared__ float smem[8][256];
    const int lane        = threadIdx.x & 31;
    const int waveInBlock = threadIdx.x >> 5;
    const int gwave       = blockIdx.x * 8 + waveInBlock;
    const int h           = gwave >> 4;          // 16 waves cover one 256-px row
    const int w0          = (gwave & 15) << 4;
    const int m           = lane & 15;
    const int halfSel     = lane >> 4;           // 0: samples 4t,4t+1  1: 4t+2,4t+3
    const int w           = w0 + m;

    // Camera: origin = -R^T T, dir = R^T (gx*tan, gy*tan, 1)
    const float R00=Rm[0],R01=Rm[1],R02=Rm[2],R10=Rm[3],R11=Rm[4],R12=Rm[5],
                R20=Rm[6],R21=Rm[7],R22=Rm[8];
    const float T0=Tm[0],T1=Tm[1],T2=Tm[2];
    const float ox = -(R00*T0 + R10*T1 + R20*T2);
    const float oy = -(R01*T0 + R11*T1 + R21*T2);
    const float oz = -(R02*T0 + R12*T1 + R22*T2);
    const float gx = fmaf((float)w, GSTEP, -1.0f) * FOV_TAN;
    const float gy = fmaf((float)h, GSTEP, -1.0f) * FOV_TAN;
    const float dx = R00*gx + R10*gy + R20;
    const float dy = R01*gx + R11*gy + R21;
    const float dz = R02*gx + R12*gy + R22;
    // local(p) = (origin + dir*depth)/half  (hoist the divisions)
    const float oxh = ox / HALFC, oyh = oy / HALFC, ozh = oz / HALFC;
    const float dxh = dx / HALFC, dyh = dy / HALFC, dzh = dz / HALFC;

    // Pass 1: inside-interval (convex cube + monotone coords => contiguous).
    int firstIn = NPTS, cnt = 0;
    for (int p = 0; p < NPTS; ++p) {
        const float depth = fmaf((float)p, DSTEP, 2.0f);
        const float lx = fmaf(dxh, depth, oxh);
        const float ly = fmaf(dyh, depth, oyh);
        const float lz = fmaf(dzh, depth, ozh);
        const int ins = (fabsf(lx) <= 1.0f) & (fabsf(ly) <= 1.0f) & (fabsf(lz) <= 1.0f);
        firstIn = min(firstIn, ins ? p : NPTS);
        cnt += ins;
    }

    // Transmittance factor: f32(1 - 0.1 + 1e-10); note f32(1 + 1e-10) == 1 exactly,
    // so outside samples contribute no factor. n jumps by at most 3 per lane sample.
    const float om  = (1.0f - 0.1f) + 1e-10f;
    const float om2 = om * om, om3 = om2 * om;
    float curT  = 1.0f;
    int   prevN = 0;
    v8f   acc   = {};

    for (int t = 0; t < 80; ++t) {
        v2f aW, bF;
        #pragma unroll
        for (int s = 0; s < 2; ++s) {
            const int p = 4 * t + 2 * halfSel + s;
            const float depth = fmaf((float)p, DSTEP, 2.0f);
            const float lx = fmaf(dxh, depth, oxh);
            const float ly = fmaf(dyh, depth, oyh);
            const float lz = fmaf(dzh, depth, ozh);
            const bool ins = (fabsf(lx) <= 1.0f) && (fabsf(ly) <= 1.0f) && (fabsf(lz) <= 1.0f);
            const int n  = min(max(p - firstIn, 0), cnt);   // #inside samples before p
            const int dn = n - prevN; prevN = n;
            curT *= (dn == 0) ? 1.0f : ((dn == 1) ? om : ((dn == 2) ? om2 : om3));
            float f = 0.0f, wgt = 0.0f;
            if (ins) {
                const float fx = ((lx + 1.0f) * 0.5f) * 255.0f;
                const float fy = ((ly + 1.0f) * 0.5f) * 255.0f;
                const float fz = ((lz + 1.0f) * 0.5f) * 255.0f;
                const float x0 = floorf(fx), y0 = floorf(fy), z0 = floorf(fz);
                const float wx = fx - x0, wy = fy - y0, wz = fz - z0;
                const int x0i = min(max((int)x0, 0), 255), x1i = min(max((int)x0 + 1, 0), 255);
                const int y0i = min(max((int)y0, 0), 255), y1i = min(max((int)y0 + 1, 0), 255);
                const int z0i = min(max((int)z0, 0), 255), z1i = min(max((int)z0 + 1, 0), 255);
                const int zo0 = z0i << 16, zo1 = z1i << 16, yo0 = y0i << 8, yo1 = y1i << 8;
                const float v000 = vol[zo0 + yo0 + x0i];
                const float v001 = vol[zo0 + yo0 + x1i];
                const float v010 = vol[zo0 + yo1 + x0i];
                const float v011 = vol[zo0 + yo1 + x1i];
                const float v100 = vol[zo1 + yo0 + x0i];
                const float v101 = vol[zo1 + yo0 + x1i];
                const float v110 = vol[zo1 + yo1 + x0i];
                const float v111 = vol[zo1 + yo1 + x1i];
                const float iwx = 1.0f - wx, iwy = 1.0f - wy, iwz = 1.0f - wz;
                float o = v000 * (iwz * iwy * iwx);
                o += v001 * (iwz * iwy * wx);
                o += v010 * (iwz * wy * iwx);
                o += v011 * (iwz * wy * wx);
                o += v100 * (wz * iwy * iwx);
                o += v101 * (wz * iwy * wx);
                o += v110 * (wz * wy * iwx);
                o += v111 * (wz * wy * wx);
                f   = o;
                wgt = 0.1f * curT;      // alpha * exclusive-prefix transmittance
            }
            if (s == 0) { aW.x = wgt; bF.x = f; } else { aW.y = wgt; bF.y = f; }
        }
        // D += A x B ; diagonal D[m][m] accumulates pixel m's  Sum w*f  (full f32)
        acc = __builtin_amdgcn_wmma_f32_16x16x4_f32(
            /*neg_a=*/false, aW, /*neg_b=*/false, bF,
            /*c_mod=*/(short)0, acc, /*reuse_a=*/false, /*reuse_b=*/false);
    }

    // Extract diagonal via LDS. D layout: VGPR r -> M=r (lanes 0-15, N=lane) and
    // M=r+8 (lanes 16-31, N=lane-16). So D[m][m]: m<8 -> (lane m, vgpr m);
    // m>=8 -> (lane m+16, vgpr m-8).
    float* my = smem[waveInBlock];
    #pragma unroll
    for (int r = 0; r < 8; ++r) my[lane * 8 + r] = acc[r];
    __syncthreads();
    if (lane < 16) {
        const int srcLane = (m < 8) ? m : (m + 16);
        const int r       = (m < 8) ? m : (m - 8);
        // output is transposed: flat index = w*256 + h
        grayOut[w * IMG + h] = my[srcLane * 8 + r];
    }
}

__global__ __launch_bounds__(256) void stats_kernel(const float* __restrict__ g,
                                                    float* __restrict__ stats)
{
    __shared__ double ssum[256], ssq[256];
    __shared__ float  smn[256], smx[256];
    const int tid = threadIdx.x;
    double s = 0.0, sq = 0.0;
    float mn = 3.0e38f, mx = -3.0e38f;
    for (int i = tid; i < IMG * IMG; i += 256) {
        const float v = g[i];
        s += (double)v; sq += (double)v * (double)v;
        mn = fminf(mn, v); mx = fmaxf(mx, v);
    }
    ssum[tid] = s; ssq[tid] = sq; smn[tid] = mn; smx[tid] = mx;
    __syncthreads();
    for (int off = 128; off > 0; off >>= 1) {
        if (tid < off) {
            ssum[tid] += ssum[tid + off];
            ssq[tid]  += ssq[tid + off];
            smn[tid]   = fminf(smn[tid], smn[tid + off]);
            smx[tid]   = fmaxf(smx[tid], smx[tid + off]);
        }
        __syncthreads();
    }
    if (tid == 0) {
        const double N    = (double)(IMG * IMG);
        const double mean = ssum[0] / N;
        double var = (ssq[0] - ssum[0] * ssum[0] / N) / (N - 1.0);
        if (var < 0.0) var = 0.0;
        const float meanf  = (float)mean;
        const float denom  = (float)sqrt(var) + 1e-8f;           // sigma(ddof=1)+EPS
        const float stdmin = (smn[0] - meanf) / denom;           // matches elementwise f32
        const float stdmax = (smx[0] - meanf) / denom;
        stats[0] = meanf;
        stats[1] = denom;
        stats[2] = stdmin;
        stats[3] = stdmax - stdmin + 1e-8f;
    }
}

__global__ __launch_bounds__(256) void norm_kernel(float* __restrict__ g,
                                                   const float* __restrict__ stats)
{
    const int i = blockIdx.x * 256 + threadIdx.x;
    const float meanf = stats[0], denom = stats[1], stdmin = stats[2], range = stats[3];
    const float sd = (g[i] - meanf) / denom;
    g[i] = (sd - stdmin + 1e-8f) / range;
}

extern "C" void kernel_launch(void* const* d_in, const int* in_sizes, int n_in,
                              void* d_out, int out_size, void* d_ws, size_t ws_size,
                              hipStream_t stream) {
    (void)in_sizes; (void)n_in; (void)out_size; (void)ws_size;
    const float* vol = (const float*)d_in[0];   // (1,1,256,256,256) f32
    const float* Rm  = (const float*)d_in[1];   // (1,3,3) f32
    const float* Tm  = (const float*)d_in[2];   // (1,3)   f32
    float* out   = (float*)d_out;               // 65536 f32, gray then normalized in place
    float* stats = (float*)d_ws;                // 4 f32

    // 65536 px / 16 px-per-wave / 8 waves-per-block = 512 blocks
    render_kernel<<<512, 256, 0, stream>>>(vol, Rm, Tm, out);
    stats_kernel<<<1, 256, 0, stream>>>(out, stats);
    norm_kernel<<<IMG * IMG / 256, 256, 0, stream>>>(out, stats);
}